// Encoder_26766236188766
// MI455X (gfx1250) — compile-verified
//
#include <hip/hip_runtime.h>
#include <hip/hip_bf16.h>

typedef __attribute__((ext_vector_type(2))) float v2f;
typedef __attribute__((ext_vector_type(8))) float v8f;

#define D 128
#define EPS 1e-5f

__device__ __forceinline__ void atomicAddF32(float* p, float v) {
    __hip_atomic_fetch_add(p, v, __ATOMIC_RELAXED, __HIP_MEMORY_SCOPE_AGENT);
}

// ---------------- degree / normalization ----------------
__global__ void k_deg_init(float* deg, int n) {
    int i = blockIdx.x * blockDim.x + threadIdx.x;
    if (i < n) deg[i] = 1.0f;  // self-loop contributes 1 to every node's degree
}

__global__ void k_deg_accum(const long long* __restrict__ dst, float* deg, int e) {
    int i = blockIdx.x * blockDim.x + threadIdx.x;
    if (i < e) atomicAddF32(&deg[dst[i]], 1.0f);
}

__global__ void k_deg_rsqrt(float* deg, int n) {
    int i = blockIdx.x * blockDim.x + threadIdx.x;
    if (i < n) deg[i] = __frsqrt_rn(deg[i]);  // deg >= 1 always
}

// ---------------- WMMA GEMM: C[N x 128] = A[N x 128] @ W[128 x 128] ----------------
// Block = 256 threads = 8 waves. Block b covers rows [16b, 16b+16); wave w covers
// cols [16w, 16w+16). One wave accumulates a 16x16 f32 tile with 32 steps of
// v_wmma_f32_16x16x4_f32. N must be a multiple of 16 (100000 = 6250*16): no
// divergence, EXEC all ones as WMMA requires.
__global__ void k_gemm_wmma(const float* __restrict__ A, const float* __restrict__ W,
                            float* __restrict__ C, int nrows) {
    const int lane = threadIdx.x & 31;
    const int wave = threadIdx.x >> 5;
    const int row0 = blockIdx.x << 4;
    const int col0 = wave << 4;
    const int m    = lane & 15;          // row (A) / col (B,C) within tile
    const int koff = (lane >> 4) << 1;   // lanes 0-15: K+{0,1}; lanes 16-31: K+{2,3}

    const float* Arow = A + (size_t)(row0 + m) * D;
    v8f acc = {};
    #pragma unroll
    for (int k0 = 0; k0 < D; k0 += 4) {
        v2f a, b;
        a.x = Arow[k0 + koff];
        a.y = Arow[k0 + koff + 1];
        b.x = W[(size_t)(k0 + koff)     * D + col0 + m];
        b.y = W[(size_t)(k0 + koff + 1) * D + col0 + m];
        acc = __builtin_amdgcn_wmma_f32_16x16x4_f32(
            /*neg_a=*/false, a, /*neg_b=*/false, b,
            /*c_mod=*/(short)0, acc, /*reuse_a=*/false, /*reuse_b=*/false);
    }
    // C/D layout: VGPR r -> M=r (lanes 0-15), M=r+8 (lanes 16-31); N = lane&15
    const int half = lane >> 4;
    #pragma unroll
    for (int r = 0; r < 8; ++r) {
        C[(size_t)(row0 + r + half * 8) * D + col0 + m] = acc[r];
    }
}

// ---------------- aggregation ----------------
// out[i,c] = h[i,c]*dinv[i]^2 + bias[c]   (self-loop term + bias)
__global__ void k_agg_init(const float* __restrict__ h, const float* __restrict__ dinv,
                           const float* __restrict__ bias, float* __restrict__ out, int n) {
    int t = blockIdx.x * blockDim.x + threadIdx.x;
    if (t >= n * D) return;
    int i = t >> 7, c = t & (D - 1);
    float dv = dinv[i];
    out[t] = h[t] * dv * dv + bias[c];
}

// one wave per edge, 4 columns (float4) per lane; atomic fadd into out[dst]
__global__ void k_edge_agg(const float* __restrict__ h, const long long* __restrict__ src,
                           const long long* __restrict__ dst, const float* __restrict__ dinv,
                           float* __restrict__ out, int e) {
    int t = blockIdx.x * blockDim.x + threadIdx.x;
    int ed = t >> 5;
    int lane = t & 31;
    if (ed >= e) return;
    long long s = src[ed], d = dst[ed];
    float norm = dinv[s] * dinv[d];
    const float4 v = *(const float4*)(h + (size_t)s * D + lane * 4);
    float* o = out + (size_t)d * D + lane * 4;
    atomicAddF32(o + 0, v.x * norm);
    atomicAddF32(o + 1, v.y * norm);
    atomicAddF32(o + 2, v.z * norm);
    atomicAddF32(o + 3, v.w * norm);
}

// ---------------- batch norm + PReLU ----------------
__global__ void k_zero(float* p, int n) {
    int i = blockIdx.x * blockDim.x + threadIdx.x;
    if (i < n) p[i] = 0.0f;
}

// stats[0:128]=sum, stats[128:256]=sumsq ; block handles 512 rows, 128 threads = 1 col each
__global__ void k_bn_stats(const float* __restrict__ h, float* __restrict__ stats, int n) {
    int c = threadIdx.x;  // 0..127
    int r0 = blockIdx.x * 512;
    int r1 = min(r0 + 512, n);
    float s = 0.0f, sq = 0.0f;
    for (int r = r0; r < r1; ++r) {
        float v = h[(size_t)r * D + c];
        s += v;
        sq += v * v;
    }
    atomicAddF32(&stats[c], s);
    atomicAddF32(&stats[D + c], sq);
}

__global__ void k_bn_finalize(const float* __restrict__ stats, const float* __restrict__ gamma,
                              const float* __restrict__ beta, float* __restrict__ ss, int n) {
    int c = threadIdx.x;
    if (c >= D) return;
    float inv_n = 1.0f / (float)n;
    float mean = stats[c] * inv_n;
    float var = fmaxf(stats[D + c] * inv_n - mean * mean, 0.0f);
    float sc = gamma[c] * __frsqrt_rn(var + EPS);
    ss[c] = sc;                      // scale
    ss[D + c] = beta[c] - mean * sc; // shift
}

__global__ void k_bn_prelu(const float* __restrict__ h, const float* __restrict__ ss,
                           const float* __restrict__ a, float* __restrict__ out, int n) {
    int t = blockIdx.x * blockDim.x + threadIdx.x;
    if (t >= n * D) return;
    int c = t & (D - 1);
    float v = fmaf(h[t], ss[c], ss[D + c]);
    float alpha = a[0];
    out[t] = v > 0.0f ? v : alpha * v;
}

// ---------------- host-side orchestration ----------------
extern "C" void kernel_launch(void* const* d_in, const int* in_sizes, int n_in,
                              void* d_out, int out_size, void* d_ws, size_t ws_size,
                              hipStream_t stream) {
    const float* x          = (const float*)d_in[0];
    const long long* ei     = (const long long*)d_in[1];  // int64 [2, E]
    const float* w0         = (const float*)d_in[2];
    const float* b0         = (const float*)d_in[3];
    const float* gamma0     = (const float*)d_in[4];
    const float* beta0      = (const float*)d_in[5];
    const float* a0         = (const float*)d_in[6];
    const float* w1         = (const float*)d_in[7];
    const float* b1         = (const float*)d_in[8];
    const float* gamma1     = (const float*)d_in[9];
    const float* beta1      = (const float*)d_in[10];
    const float* a1         = (const float*)d_in[11];
    float* out              = (float*)d_out;

    const int N = in_sizes[0] / D;
    const int E = in_sizes[1] / 2;
    const long long* src = ei;
    const long long* dst = ei + E;

    // workspace layout
    float* bufA  = (float*)d_ws;                 // N*128
    float* bufB  = bufA + (size_t)N * D;         // N*128
    float* dinv  = bufB + (size_t)N * D;         // N
    float* stats = dinv + N;                     // 256
    float* ss    = stats + 2 * D;                // 256 (scale, shift)

    const int T = 256;
    const int gN    = (N + T - 1) / T;
    const int gE    = (E + T - 1) / T;
    const int gNE   = ((size_t)N * D + T - 1) / T;       // elementwise over N*128
    const int gEdge = (int)(((size_t)E * 32 + T - 1) / T);  // 1 wave per edge
    const int gGemm = N / 16;                             // 16-row strips
    const int gStat = (N + 511) / 512;

    // --- symmetric normalization: dinv = rsqrt(deg), deg = 1 + indegree ---
    k_deg_init<<<gN, T, 0, stream>>>(dinv, N);
    k_deg_accum<<<gE, T, 0, stream>>>(dst, dinv, E);
    k_deg_rsqrt<<<gN, T, 0, stream>>>(dinv, N);

    // ===================== layer 0 =====================
    k_gemm_wmma<<<gGemm, T, 0, stream>>>(x, w0, bufA, N);
    k_agg_init<<<gNE, T, 0, stream>>>(bufA, dinv, b0, bufB, N);
    k_edge_agg<<<gEdge, T, 0, stream>>>(bufA, src, dst, dinv, bufB, E);
    k_zero<<<1, 2 * D, 0, stream>>>(stats, 2 * D);
    k_bn_stats<<<gStat, D, 0, stream>>>(bufB, stats, N);
    k_bn_finalize<<<1, D, 0, stream>>>(stats, gamma0, beta0, ss, N);
    k_bn_prelu<<<gNE, T, 0, stream>>>(bufB, ss, a0, bufA, N);

    // ===================== layer 1 =====================
    k_gemm_wmma<<<gGemm, T, 0, stream>>>(bufA, w1, bufB, N);
    k_agg_init<<<gNE, T, 0, stream>>>(bufB, dinv, b1, out, N);
    k_edge_agg<<<gEdge, T, 0, stream>>>(bufB, src, dst, dinv, out, E);
    k_zero<<<1, 2 * D, 0, stream>>>(stats, 2 * D);
    k_bn_stats<<<gStat, D, 0, stream>>>(out, stats, N);
    k_bn_finalize<<<1, D, 0, stream>>>(stats, gamma1, beta1, ss, N);
    k_bn_prelu<<<gNE, T, 0, stream>>>(out, ss, a1, out, N);
}